// GInvariants_52725018525982
// MI455X (gfx1250) — compile-verified
//
#include <hip/hip_runtime.h>
#include <hip/hip_bf16.h>

typedef __attribute__((ext_vector_type(2))) float v2f;
typedef __attribute__((ext_vector_type(8))) float v8f;

// out[b,i,j] = sum_d p[b,i,d] * eta[d] * p[b,j,d],  eta = (+1,-1,-1,-1)
// B=128, N=512, D=4.  One V_WMMA_F32_16X16X4_F32 per 16x16 output tile.
// Block = 256 threads (8 waves) handles one (b, i-tile); each wave does 4 j-tiles.
__global__ __launch_bounds__(256) void GInvariants_52725018525982_kernel(
    const float* __restrict__ p, float* __restrict__ out) {
  constexpr int N = 512;

  const int lane = threadIdx.x & 31;
  const int wave = threadIdx.x >> 5;
  const int b    = blockIdx.x >> 5;   // batch
  const int ti   = blockIdx.x & 31;   // i-tile (32 tiles of 16 rows)

  const int half = lane >> 4;         // 0: K={0,1}, 1: K={2,3}
  const int mn   = lane & 15;         // A row / B col within tile
  const int k0   = half << 1;

  const float* pb = p + (size_t)b * (N * 4);

  // ---- A fragment: rows ti*16 + mn of p (16x4, f32 -> 2 VGPRs) ----
  const int arow = ti * 16 + mn;
  const v2f a = *(const v2f*)(pb + arow * 4 + k0);

  // Minkowski signs for this lane's two K slots: eta[k0], eta[k0+1]
  const float s0 = (half == 0) ? 1.0f : -1.0f;  // eta[0]=+1, eta[2]=-1
  const float s1 = -1.0f;                       // eta[1]=eta[3]=-1

  float* outb = out + (size_t)b * N * N + (size_t)(ti * 16 + 8 * half) * N;

  #pragma unroll
  for (int t = 0; t < 4; ++t) {
    const int tj   = (wave << 2) + t;
    const int brow = tj * 16 + mn;

    // ---- B fragment: column n = mn holds eta-weighted row of p ----
    v2f bb = *(const v2f*)(pb + brow * 4 + k0);
    bb.x *= s0;
    bb.y *= s1;

    v8f c = {0.f, 0.f, 0.f, 0.f, 0.f, 0.f, 0.f, 0.f};
    // 8 args: (neg_a, A, neg_b, B, c_mod, C, reuse_a, reuse_b)
    c = __builtin_amdgcn_wmma_f32_16x16x4_f32(
        false, a, false, bb, (short)0, c, false, false);

    // ---- store D: lane l, VGPR v -> row m = v + 8*half, col n = mn ----
    float* o = outb + tj * 16 + mn;
    #pragma unroll
    for (int v = 0; v < 8; ++v) {
      __builtin_nontemporal_store(c[v], o + (size_t)v * N);
    }
  }
}

extern "C" void kernel_launch(void* const* d_in, const int* in_sizes, int n_in,
                              void* d_out, int out_size, void* d_ws, size_t ws_size,
                              hipStream_t stream) {
  (void)in_sizes; (void)n_in; (void)d_ws; (void)ws_size; (void)out_size;
  const float* p = (const float*)d_in[0];
  float* out = (float*)d_out;
  // grid: 128 batches * 32 i-tiles; block: 256 threads = 8 waves
  dim3 grid(128 * 32), block(256);
  GInvariants_52725018525982_kernel<<<grid, block, 0, stream>>>(p, out);
}